// GraphAttention_6468220748601
// MI455X (gfx1250) — compile-verified
//
#include <hip/hip_runtime.h>
#include <math.h>

// ---------------------------------------------------------------------------
// Problem constants (match reference)
// ---------------------------------------------------------------------------
constexpr int C0 = 64, C1 = 32, RBF = 64, HID = 128, NW = 224; // 64+64+32+32+32
constexpr int HEADS = 8, ALPHA_DIM = 32, HEAD_S = 16, HEAD_V = 8;
constexpr int TPS = C0 + C1;          // 96
constexpr int TPV = C0 + 2 * C1;      // 128
constexpr int FS = HEADS * (ALPHA_DIM + HEAD_S); // 384
constexpr int VV = HEADS * HEAD_V;    // 64

typedef float v2f __attribute__((ext_vector_type(2)));
typedef float v8f __attribute__((ext_vector_type(8)));

// ---------------------------------------------------------------------------
// Weight pre-pack: B fragments in lane order so each lane's two k-values are
// contiguous ->  one coalesced global_load_b64 per WMMA instead of 2x b32.
//   pack[( tile*(K/4) + kq )*32 + lane] = { B[kq*4+kh][tile*16+c],
//                                           B[kq*4+kh+1][tile*16+c] }
//   with c = lane&15, kh = (lane>>4)*2  (CDNA5 ISA 7.12.2 B layout, wave32)
// ---------------------------------------------------------------------------
__global__ void pack_b_kernel(const float* __restrict__ W, float* __restrict__ out,
                              int K, int N) {
  int i = blockIdx.x * blockDim.x + threadIdx.x;
  int total = (K * N) >> 1;              // number of v2f fragments
  int stride = gridDim.x * blockDim.x;
  for (int j = i; j < total; j += stride) {
    int lane = j & 31;
    int rest = j >> 5;
    int kq = rest % (K >> 2);
    int tile = rest / (K >> 2);
    int c = lane & 15;
    int kh = (lane >> 4) << 1;
    int kb = kq * 4 + kh;
    out[2 * j + 0] = W[(size_t)kb * N + tile * 16 + c];
    out[2 * j + 1] = W[(size_t)(kb + 1) * N + tile * 16 + c];
  }
}

// ---------------------------------------------------------------------------
// fp32 WMMA 16x16 tile from LDS A (row-major, lda) and packed global B.
// ---------------------------------------------------------------------------
__device__ __forceinline__ v8f wmma_tile_f32p(const float* __restrict__ A, int lda,
                                              const float* __restrict__ Bp, int K,
                                              int tile, int lane) {
  v8f acc = {0.f, 0.f, 0.f, 0.f, 0.f, 0.f, 0.f, 0.f};
  const int r  = lane & 15;
  const int kh = (lane >> 4) << 1;            // 0 or 2
  const float* __restrict__ arow = A + r * lda + kh;
  const v2f* __restrict__ bp =
      (const v2f*)Bp + (size_t)tile * (K >> 2) * 32 + lane;
  for (int kq = 0; kq < (K >> 2); ++kq) {
    v2f a; a.x = arow[kq * 4]; a.y = arow[kq * 4 + 1];
    v2f b = bp[kq * 32];
    acc = __builtin_amdgcn_wmma_f32_16x16x4_f32(false, a, false, b, (short)0, acc,
                                                false, false);
  }
  return acc;
}

__device__ __forceinline__ float siluf(float x) { return x / (1.f + expf(-x)); }

__device__ __forceinline__ unsigned seg_encode(float x) {
  unsigned b = __float_as_uint(x);
  return (b & 0x80000000u) ? ~b : (b | 0x80000000u);
}
__device__ __forceinline__ float seg_decode(unsigned key) {
  if (key == 0u) return 0.f;                       // untouched -> no edges -> 0
  unsigned b = (key & 0x80000000u) ? (key ^ 0x80000000u) : ~key;
  float f = __uint_as_float(b);
  return isfinite(f) ? f : 0.f;
}

// ---------------------------------------------------------------------------
// Kernel A: node transforms  hs_* = x_s @ Ws_*,  hv_*[n][d][i] = sum_c x_v*Wv
// ---------------------------------------------------------------------------
__global__ void node_transform_kernel(const float* __restrict__ xs,
                                      const float* __restrict__ xv,
                                      const float* __restrict__ Ws_src,
                                      const float* __restrict__ Ws_dst,
                                      const float* __restrict__ Wv_src,
                                      const float* __restrict__ Wv_dst,
                                      float* __restrict__ hs_src,
                                      float* __restrict__ hs_dst,
                                      float* __restrict__ hv_src,
                                      float* __restrict__ hv_dst, int n) {
  int node = blockIdx.x;
  if (node >= n) return;
  __shared__ float s_xs[C0];
  __shared__ float s_xv[C1 * 3];
  int t = threadIdx.x;
  if (t < C0) s_xs[t] = xs[(size_t)node * C0 + t];
  if (t >= 64 && t < 160) s_xv[t - 64] = xv[(size_t)node * (C1 * 3) + (t - 64)];
  __syncthreads();
  if (t < C0) {
    float a0 = 0.f, a1 = 0.f;
    for (int c = 0; c < C0; ++c) {
      float x = s_xs[c];
      a0 += x * Ws_src[c * C0 + t];
      a1 += x * Ws_dst[c * C0 + t];
    }
    hs_src[(size_t)node * C0 + t] = a0;
    hs_dst[(size_t)node * C0 + t] = a1;
  } else {
    int o = t - 64;                     // 0..191
    int mat = o / 96, rem = o % 96;
    int d = rem / 3, i = rem % 3;
    const float* Wv = mat ? Wv_dst : Wv_src;
    float acc = 0.f;
    for (int c = 0; c < C1; ++c) acc += s_xv[c * 3 + i] * Wv[c * C1 + d];
    (mat ? hv_dst : hv_src)[(size_t)node * (C1 * 3) + rem] = acc;
  }
}

// ---------------------------------------------------------------------------
// Init: zero segment-max keys, denominators and aggregation buffers
// ---------------------------------------------------------------------------
__global__ void init_kernel(unsigned* __restrict__ segmax, float* __restrict__ denom,
                            float* __restrict__ agg_s, float* __restrict__ agg_v, int n) {
  int i = blockIdx.x * blockDim.x + threadIdx.x;
  int stride = gridDim.x * blockDim.x;
  for (int j = i; j < n * HEADS; j += stride) { segmax[j] = 0u; denom[j] = 0.f; }
  for (int j = i; j < n * HEADS * HEAD_S; j += stride) agg_s[j] = 0.f;
  for (int j = i; j < n * VV * 3; j += stride) agg_v[j] = 0.f;
}

// ---------------------------------------------------------------------------
// Kernel B: full per-edge pipeline, 16 edges per 256-thread block (8 waves).
// LDS layout (floats), padded leading dims for bank-conflict freedom:
// ---------------------------------------------------------------------------
constexpr int MS_LD  = 65;   // ms   16x65
constexpr int W_LD   = 225;  // w    16x225
constexpr int ES_LD  = 65;   // es tile (region A overlay)
constexpr int TPV_E  = 385;  // tp_v per-edge stride (3*128 + pad)
constexpr int HID_LD = 129;  // hid  (region B overlay)
constexpr int TPS_LD = 97;   // tp_s (region B overlay)

constexpr int OFF_MS  = 0;                       // 1040
constexpr int OFF_MV  = OFF_MS + 16 * MS_LD;     // +1536 (16 x 96, [d*3+i])
constexpr int OFF_SH  = OFF_MV + 16 * 96;        // +64   (16 x 4)
constexpr int OFF_W   = OFF_SH + 16 * 4;         // +3600
constexpr int OFF_RA  = OFF_W + 16 * W_LD;       // +6160 (es | tp_v)
constexpr int OFF_RB  = OFF_RA + 16 * TPV_E;     // +2064 (hid | tp_s)
constexpr int OFF_LOG = OFF_RB + 16 * HID_LD;    // +128
constexpr int SMEM_SZ = OFF_LOG + 16 * HEADS;    // 14592 floats = 58368 B

__global__ void __launch_bounds__(256, 1)
edge_kernel(const float* __restrict__ edge_vec,
            const float* __restrict__ edge_scalars,
            const int* __restrict__ edge_src,
            const int* __restrict__ edge_dst,
            const float* __restrict__ hs_src, const float* __restrict__ hs_dst,
            const float* __restrict__ hv_src, const float* __restrict__ hv_dst,
            const float* __restrict__ pW1, const float* __restrict__ rad_b1,
            const float* __restrict__ pW2, const float* __restrict__ rad_b2,
            const float* __restrict__ pWs, const float* __restrict__ lin_s_b,
            const float* __restrict__ pWv, const float* __restrict__ alpha_dotp,
            float* __restrict__ logits_g, float* __restrict__ val_s_g,
            float* __restrict__ val_v_g, unsigned* __restrict__ segmax_g,
            int Etot) {
  __shared__ float smem[SMEM_SZ];
  __shared__ int s_src[16], s_dst[16];

  const int t = threadIdx.x;
  const int lane = t & 31;
  const int wave = t >> 5;          // 0..7
  const int e0 = blockIdx.x * 16;

  // ---- stage 0a: per-edge scalars (unit vector, indices) -------------------
  if (t < 16) {
    int e = e0 + t;
    int ec = (e < Etot) ? e : 0;
    s_src[t] = edge_src[ec];
    s_dst[t] = edge_dst[ec];
    float vx = edge_vec[(size_t)ec * 3 + 0];
    float vy = edge_vec[(size_t)ec * 3 + 1];
    float vz = edge_vec[(size_t)ec * 3 + 2];
    float inv = 1.7320508075688772f / (sqrtf(vx * vx + vy * vy + vz * vz) + 1e-9f);
    smem[OFF_SH + t * 4 + 0] = vx * inv;   // sh1 = sqrt(3)*unit
    smem[OFF_SH + t * 4 + 1] = vy * inv;
    smem[OFF_SH + t * 4 + 2] = vz * inv;
  }
  if (t < 16 * HEADS) smem[OFF_LOG + t] = 0.f;
  __syncthreads();

  // ---- stage 0b: gather ms, mv; stage edge_scalars tile --------------------
  for (int idx = t; idx < 16 * C0; idx += 256) {
    int row = idx >> 6, c = idx & 63;
    int s = s_src[row], d = s_dst[row];
    smem[OFF_MS + row * MS_LD + c] =
        hs_src[(size_t)s * C0 + c] + hs_dst[(size_t)d * C0 + c];
    int e = e0 + row; if (e >= Etot) e = 0;
    smem[OFF_RA + row * ES_LD + c] = edge_scalars[(size_t)e * RBF + c];
  }
  for (int idx = t; idx < 16 * 96; idx += 256) {
    int row = idx / 96, c = idx % 96;
    smem[OFF_MV + row * 96 + c] =
        hv_src[(size_t)s_src[row] * 96 + c] + hv_dst[(size_t)s_dst[row] * 96 + c];
  }
  __syncthreads();

  // ---- stage 1: hid = silu(es @ rad_W1 + b1)   (WMMA, 8 waves x 1 tile) ----
  {
    int n0 = wave * 16;               // HID = 128 = 8*16
    v8f acc = wmma_tile_f32p(&smem[OFF_RA], ES_LD, pW1, RBF, wave, lane);
    int col = lane & 15, rowo = (lane >> 4) << 3;
    for (int r = 0; r < 8; ++r) {
      float x = acc[r] + rad_b1[n0 + col];
      smem[OFF_RB + (rowo + r) * HID_LD + n0 + col] = siluf(x);
    }
  }
  __syncthreads();

  // ---- stage 2: w = hid @ rad_W2 + b2   (WMMA, 14 tiles over 8 waves) ------
  for (int T = wave; T < NW / 16; T += 8) {
    int n0 = T * 16;
    v8f acc = wmma_tile_f32p(&smem[OFF_RB], HID_LD, pW2, HID, T, lane);
    int col = lane & 15, rowo = (lane >> 4) << 3;
    for (int r = 0; r < 8; ++r)
      smem[OFF_W + (rowo + r) * W_LD + n0 + col] = acc[r] + rad_b2[n0 + col];
  }
  __syncthreads();

  // ---- stage 3: tensor products tp_s (16x96), tp_v (16x3x128) (VALU) -------
  {
    int row = t >> 4, j = t & 15;
    const float* wr = &smem[OFF_W + row * W_LD];
    const float* msr = &smem[OFF_MS + row * MS_LD];
    const float* mvr = &smem[OFF_MV + row * 96];
    float shx = smem[OFF_SH + row * 4 + 0];
    float shy = smem[OFF_SH + row * 4 + 1];
    float shz = smem[OFF_SH + row * 4 + 2];
    float* tps = &smem[OFF_RB + row * TPS_LD];     // overlays dead hid
    float* tpv = &smem[OFF_RA + row * TPV_E];      // overlays dead es
    for (int c = j; c < C0; c += 16) {
      float m = msr[c];
      tps[c] = wr[c] * m;                          // wA * ms
      float v = wr[C0 + c] * m;                    // wB * ms
      tpv[0 * TPV + c] = v * shx;
      tpv[1 * TPV + c] = v * shy;
      tpv[2 * TPV + c] = v * shz;
    }
    for (int d = j; d < C1; d += 16) {
      float ax = mvr[d * 3 + 0], ay = mvr[d * 3 + 1], az = mvr[d * 3 + 2];
      float dot = ax * shx + ay * shy + az * shz;
      tps[C0 + d] = wr[2 * C0 + C1 + d] * dot * 0.5773502691896258f; // wD/sqrt3
      float wc = wr[2 * C0 + d];                                     // wC
      tpv[0 * TPV + C0 + d] = wc * ax;
      tpv[1 * TPV + C0 + d] = wc * ay;
      tpv[2 * TPV + C0 + d] = wc * az;
      float we = wr[2 * C0 + 2 * C1 + d] * 0.7071067811865475f;      // wE/sqrt2
      tpv[0 * TPV + C0 + C1 + d] = we * (ay * shz - az * shy);
      tpv[1 * TPV + C0 + C1 + d] = we * (az * shx - ax * shz);
      tpv[2 * TPV + C0 + C1 + d] = we * (ax * shy - ay * shx);
    }
  }
  __syncthreads();

  // ---- stage 4: lin_s (24 tiles) + lin_v (12 tiles) over 8 waves (WMMA) ----
  for (int T = wave; T < 36; T += 8) {
    if (T < 24) {                                   // f_s = tp_s @ lin_s_W + b
      int n0 = T * 16;
      v8f acc = wmma_tile_f32p(&smem[OFF_RB], TPS_LD, pWs, TPS, T, lane);
      int col = lane & 15, rowo = (lane >> 4) << 3;
      for (int r = 0; r < 8; ++r) {
        int row = rowo + r;
        int cg = n0 + col;
        float x = acc[r] + lin_s_b[cg];
        int h = cg / (ALPHA_DIM + HEAD_S);
        int q = cg % (ALPHA_DIM + HEAD_S);
        int e = e0 + row;
        if (q < ALPHA_DIM) {                        // alpha path -> logits
          float sg = 1.f / (1.f + expf(-x));
          float slr = 0.6f * x + 0.4f * x * (2.f * sg - 1.f); // a = 0.2
          atomicAdd(&smem[OFF_LOG + row * HEADS + h],
                    slr * alpha_dotp[h * ALPHA_DIM + q]);
        } else if (e < Etot) {                      // value path
          val_s_g[(size_t)e * (HEADS * HEAD_S) + h * HEAD_S + (q - ALPHA_DIM)] = x;
        }
      }
    } else {                                        // val_v = tp_v @ lin_v_W
      int tv = T - 24;
      int i = tv >> 2;                              // spatial component
      int n0 = (tv & 3) * 16;
      v8f acc = wmma_tile_f32p(&smem[OFF_RA + i * TPV], TPV_E, pWv, TPV,
                               (tv & 3), lane);
      int col = lane & 15, rowo = (lane >> 4) << 3;
      for (int r = 0; r < 8; ++r) {
        int row = rowo + r;
        int e = e0 + row;
        if (e < Etot)
          val_v_g[(size_t)e * (VV * 3) + (n0 + col) * 3 + i] = acc[r];
      }
    }
  }
  __syncthreads();

  // ---- stage 5: emit logits + segment max atomics --------------------------
  if (t < 16 * HEADS) {
    int row = t >> 3, h = t & 7;
    int e = e0 + row;
    if (e < Etot) {
      float L = smem[OFF_LOG + row * HEADS + h];
      logits_g[(size_t)e * HEADS + h] = L;
      atomicMax(&segmax_g[(unsigned)s_dst[row] * HEADS + h], seg_encode(L));
    }
  }
}

// ---------------------------------------------------------------------------
// Kernel D: softmax denominators per (dst node, head)
// ---------------------------------------------------------------------------
__global__ void denom_kernel(const float* __restrict__ logits,
                             const int* __restrict__ dst,
                             const unsigned* __restrict__ segmax,
                             float* __restrict__ denom, int E) {
  int i = blockIdx.x * blockDim.x + threadIdx.x;
  if (i >= E * HEADS) return;
  int e = i >> 3, h = i & 7;
  int d = dst[e];
  float m = seg_decode(segmax[d * HEADS + h]);
  atomicAdd(&denom[d * HEADS + h], expf(logits[i] - m));
}

// ---------------------------------------------------------------------------
// Kernel E: attention-weighted scatter-add of val_s / val_v
// ---------------------------------------------------------------------------
__global__ void scatter_kernel(const float* __restrict__ logits,
                               const unsigned* __restrict__ segmax,
                               const float* __restrict__ denom,
                               const float* __restrict__ val_s,
                               const float* __restrict__ val_v,
                               const int* __restrict__ dst,
                               float* __restrict__ agg_s, float* __restrict__ agg_v,
                               int E) {
  long long i = (long long)blockIdx.x * blockDim.x + threadIdx.x;
  long long total = (long long)E * 320;          // 128 scalar + 192 vector slots
  if (i >= total) return;
  int e = (int)(i / 320);
  int u = (int)(i % 320);
  int d = dst[e];
  int h = (u < 128) ? (u >> 4) : ((u - 128) / 24);   // 24 = HEAD_V*3
  float m = seg_decode(segmax[d * HEADS + h]);
  float den = denom[d * HEADS + h] + 1e-9f;
  float attn = expf(logits[(size_t)e * HEADS + h] - m) / den;
  if (u < 128) {
    atomicAdd(&agg_s[(size_t)d * 128 + u], attn * val_s[(size_t)e * 128 + u]);
  } else {
    int o = u - 128;
    atomicAdd(&agg_v[(size_t)d * 192 + o], attn * val_v[(size_t)e * 192 + o]);
  }
}

// ---------------------------------------------------------------------------
// Kernel F: output projections
// ---------------------------------------------------------------------------
__global__ void out_kernel(const float* __restrict__ agg_s,
                           const float* __restrict__ agg_v,
                           const float* __restrict__ proj_s_W,
                           const float* __restrict__ proj_s_b,
                           const float* __restrict__ proj_v_W,
                           float* __restrict__ out, int n) {
  int node = blockIdx.x;
  if (node >= n) return;
  __shared__ float s[320];
  int t = threadIdx.x;   // blockDim = 160
  for (int j = t; j < 128; j += 160) s[j] = agg_s[(size_t)node * 128 + j];
  for (int j = t; j < 192; j += 160) s[128 + j] = agg_v[(size_t)node * 192 + j];
  __syncthreads();
  if (t < C0) {
    float a = proj_s_b[t];
    for (int c = 0; c < 128; ++c) a += s[c] * proj_s_W[c * C0 + t];
    out[(size_t)node * C0 + t] = a;
  } else if (t < 160) {
    int o = t - 64;                       // 0..95
    int d = o / 3, i = o % 3;
    float a = 0.f;
    for (int c = 0; c < VV; ++c) a += s[128 + c * 3 + i] * proj_v_W[c * C1 + d];
    out[(size_t)n * C0 + (size_t)node * 96 + o] = a;
  }
}

// ---------------------------------------------------------------------------
// Host launcher.  Workspace: ~237 MB fp32 scratch + 320 KB packed weights.
// ---------------------------------------------------------------------------
extern "C" void kernel_launch(void* const* d_in, const int* in_sizes, int n_in,
                              void* d_out, int out_size, void* d_ws, size_t ws_size,
                              hipStream_t stream) {
  const float* x_s          = (const float*)d_in[0];
  const float* x_v          = (const float*)d_in[1];
  const float* edge_vec     = (const float*)d_in[2];
  const float* edge_scalars = (const float*)d_in[3];
  const int*   edge_src     = (const int*)d_in[4];
  const int*   edge_dst     = (const int*)d_in[5];
  const float* Ws_src       = (const float*)d_in[6];
  const float* Ws_dst       = (const float*)d_in[7];
  const float* Wv_src       = (const float*)d_in[8];
  const float* Wv_dst       = (const float*)d_in[9];
  const float* rad_W1       = (const float*)d_in[10];
  const float* rad_b1       = (const float*)d_in[11];
  const float* rad_W2       = (const float*)d_in[12];
  const float* rad_b2       = (const float*)d_in[13];
  const float* lin_s_W      = (const float*)d_in[14];
  const float* lin_s_b      = (const float*)d_in[15];
  const float* lin_v_W      = (const float*)d_in[16];
  const float* alpha_dot    = (const float*)d_in[17];
  const float* proj_s_W     = (const float*)d_in[18];
  const float* proj_s_b     = (const float*)d_in[19];
  const float* proj_v_W     = (const float*)d_in[20];

  const int n = in_sizes[0] / C0;     // 10000
  const int E = in_sizes[4];          // 160000

  float* ws = (float*)d_ws;
  size_t off = 0;
  float* pW1 = ws + off; off += (size_t)RBF * HID;  // 8192  (packed rad_W1)
  float* pW2 = ws + off; off += (size_t)HID * NW;   // 28672 (packed rad_W2)
  float* pWs = ws + off; off += (size_t)TPS * FS;   // 36864 (packed lin_s_W)
  float* pWv = ws + off; off += (size_t)TPV * VV;   // 8192  (packed lin_v_W)
  float* hs_src  = ws + off; off += (size_t)n * C0;
  float* hs_dst  = ws + off; off += (size_t)n * C0;
  float* hv_src  = ws + off; off += (size_t)n * C1 * 3;
  float* hv_dst  = ws + off; off += (size_t)n * C1 * 3;
  float* logits  = ws + off; off += (size_t)E * HEADS;
  float* val_s   = ws + off; off += (size_t)E * HEADS * HEAD_S;
  float* val_v   = ws + off; off += (size_t)E * VV * 3;
  unsigned* segm = (unsigned*)(ws + off); off += (size_t)n * HEADS;
  float* denom   = ws + off; off += (size_t)n * HEADS;
  float* agg_s   = ws + off; off += (size_t)n * HEADS * HEAD_S;
  float* agg_v   = ws + off; off += (size_t)n * VV * 3;
  (void)ws_size; (void)n_in; (void)out_size;

  pack_b_kernel<<<16, 256, 0, stream>>>(rad_W1, pW1, RBF, HID);
  pack_b_kernel<<<56, 256, 0, stream>>>(rad_W2, pW2, HID, NW);
  pack_b_kernel<<<72, 256, 0, stream>>>(lin_s_W, pWs, TPS, FS);
  pack_b_kernel<<<16, 256, 0, stream>>>(lin_v_W, pWv, TPV, VV);

  node_transform_kernel<<<n, 256, 0, stream>>>(x_s, x_v, Ws_src, Ws_dst, Wv_src,
                                               Wv_dst, hs_src, hs_dst, hv_src,
                                               hv_dst, n);
  init_kernel<<<2048, 256, 0, stream>>>(segm, denom, agg_s, agg_v, n);
  edge_kernel<<<(E + 15) / 16, 256, 0, stream>>>(
      edge_vec, edge_scalars, edge_src, edge_dst, hs_src, hs_dst, hv_src, hv_dst,
      pW1, rad_b1, pW2, rad_b2, pWs, lin_s_b, pWv, alpha_dot,
      logits, val_s, val_v, segm, E);
  denom_kernel<<<(E * HEADS + 255) / 256, 256, 0, stream>>>(logits, edge_dst, segm,
                                                            denom, E);
  long long tot = (long long)E * 320;
  scatter_kernel<<<(int)((tot + 255) / 256), 256, 0, stream>>>(
      logits, segm, denom, val_s, val_v, edge_dst, agg_s, agg_v, E);
  out_kernel<<<n, 160, 0, stream>>>(agg_s, agg_v, proj_s_W, proj_s_b, proj_v_W,
                                    (float*)d_out, n);
}